// GraphSAGE_45122926412013
// MI455X (gfx1250) — compile-verified
//
#include <hip/hip_runtime.h>

#define N_NODES   50000
#define N_EDGES   600000
#define N_GRAPHS  500
#define HID       128
#define N_LAYERS  3
#define N_CLASSES 10

typedef float v2f __attribute__((ext_vector_type(2)));
typedef float v8f __attribute__((ext_vector_type(8)));

// ---------------------------------------------------------------- utilities
__global__ void k_copy(const float* __restrict__ src, float* __restrict__ dst, int n) {
    int i = blockIdx.x * blockDim.x + threadIdx.x;
    if (i < n) dst[i] = src[i];
}

__global__ void k_zero(float* __restrict__ p, int n) {
    int i = blockIdx.x * blockDim.x + threadIdx.x;
    if (i < n) p[i] = 0.0f;
}

// count in-degree per node (edges constant across layers -> run once)
__global__ void k_count(const int* __restrict__ ei, float* __restrict__ cnt) {
    int e = blockIdx.x * blockDim.x + threadIdx.x;
    if (e < N_EDGES) atomicAdd(&cnt[ei[N_EDGES + e]], 1.0f);
}

__global__ void k_inv(float* __restrict__ cnt) {
    int i = blockIdx.x * blockDim.x + threadIdx.x;
    if (i < N_NODES) cnt[i] = 1.0f / fmaxf(cnt[i], 1.0f);
}

// ------------------------------------------------------- edge message scatter
// one 32-lane group per edge; each lane moves 4 contiguous floats
__global__ __launch_bounds__(256) void k_scatter(const float* __restrict__ x,
                                                 const int* __restrict__ ei,
                                                 float* __restrict__ msg) {
    int t = blockIdx.x * blockDim.x + threadIdx.x;
    int e = t >> 5;
    int f = (t & 31) * 4;
    if (e >= N_EDGES) return;
    int s = ei[e];
    int d = ei[N_EDGES + e];
    const float4 v = *(const float4*)(x + (size_t)s * HID + f);
    float* mp = msg + (size_t)d * HID + f;
    atomicAdd(mp + 0, v.x);
    atomicAdd(mp + 1, v.y);
    atomicAdd(mp + 2, v.z);
    atomicAdd(mp + 3, v.w);
}

// ---------------------------------------------------- SAGE layer GEMM (WMMA)
// x_out[m,n] = relu( mean[m,:] @ Wl[n,:] + x[m,:] @ Wr[n,:] + bl[n] ), in-place
// block = 8 waves; block handles one 16-row slab, wave w handles column tile w
__global__ __launch_bounds__(256) void k_sage(float* __restrict__ x,
                                              const float* __restrict__ msg,
                                              const float* __restrict__ inv,
                                              const float* __restrict__ Wl,
                                              const float* __restrict__ bl,
                                              const float* __restrict__ Wr) {
    __shared__ float s_mean[16][HID];
    __shared__ float s_x[16][HID];

    const int rowBase = blockIdx.x * 16;
    const int tid = threadIdx.x;

    // snapshot this slab's mean and x rows into LDS (enables in-place update)
    for (int i = tid; i < 16 * HID; i += 256) {
        int r = i >> 7, c = i & (HID - 1);
        float ic = inv[rowBase + r];
        s_mean[r][c] = msg[(size_t)(rowBase + r) * HID + c] * ic;
        s_x[r][c]    = x[(size_t)(rowBase + r) * HID + c];
    }
    __syncthreads();

    const int lane = tid & 31;
    const int nb   = tid >> 5;       // column tile 0..7
    const int half = lane >> 4;      // 0: K even pair base 0, 1: base 2
    const int lm   = lane & 15;

    const int ncol = nb * 16 + lm;
    const float* __restrict__ wlrow = Wl + (size_t)ncol * HID;  // B[k][n] = W[n][k]
    const float* __restrict__ wrrow = Wr + (size_t)ncol * HID;

    v8f acc = {};
    for (int k0 = 0; k0 < HID; k0 += 4) {
        const int kk = k0 + 2 * half;
        // A tile: lane holds row lm, K = kk, kk+1 (contiguous pair)
        v2f a1 = { s_mean[lm][kk], s_mean[lm][kk + 1] };
        v2f b1 = { wlrow[kk],      wlrow[kk + 1] };
        acc = __builtin_amdgcn_wmma_f32_16x16x4_f32(false, a1, false, b1,
                                                    (short)0, acc, false, false);
        v2f a2 = { s_x[lm][kk], s_x[lm][kk + 1] };
        v2f b2 = { wrrow[kk],   wrrow[kk + 1] };
        acc = __builtin_amdgcn_wmma_f32_16x16x4_f32(false, a2, false, b2,
                                                    (short)0, acc, false, false);
    }

    // D layout: vgpr r -> M = r + 8*half, N = lm
    const float bias = bl[ncol];
    for (int r = 0; r < 8; ++r) {
        int row = rowBase + r + 8 * half;
        float v = acc[r] + bias;
        x[(size_t)row * HID + ncol] = v > 0.0f ? v : 0.0f;
    }
}

// -------------------------------------------------------------- graph pooling
__global__ __launch_bounds__(256) void k_pool(const float* __restrict__ x,
                                              const int* __restrict__ batch,
                                              float* __restrict__ g) {
    int t = blockIdx.x * blockDim.x + threadIdx.x;
    if (t >= N_NODES * HID) return;
    int n = t >> 7, f = t & (HID - 1);
    atomicAdd(&g[(size_t)batch[n] * HID + f], x[t]);
}

// ------------------------------------------------- MLP head + log-softmax
__global__ __launch_bounds__(128) void k_head(const float* __restrict__ g,
                                              const float* __restrict__ W1,
                                              const float* __restrict__ b1,
                                              const float* __restrict__ W2,
                                              const float* __restrict__ b2,
                                              float* __restrict__ out) {
    __shared__ float sg[HID];
    __shared__ float sh[HID];
    __shared__ float sl[N_CLASSES];
    const int gi = blockIdx.x;
    const int j  = threadIdx.x;

    float v = g[(size_t)gi * HID + j];
    sg[j] = v > 0.0f ? v : 0.0f;          // relu on pooled features
    __syncthreads();

    float acc = b1[j];
    const float* __restrict__ w = W1 + (size_t)j * HID;
    for (int k = 0; k < HID; ++k) acc += sg[k] * w[k];
    sh[j] = acc > 0.0f ? acc : 0.0f;
    __syncthreads();

    if (j < N_CLASSES) {
        float a = b2[j];
        const float* __restrict__ w2 = W2 + (size_t)j * HID;
        for (int k = 0; k < HID; ++k) a += sh[k] * w2[k];
        sl[j] = a;
    }
    __syncthreads();

    if (j == 0) {
        float mx = sl[0];
        for (int c = 1; c < N_CLASSES; ++c) mx = fmaxf(mx, sl[c]);
        float s = 0.0f;
        for (int c = 0; c < N_CLASSES; ++c) s += __expf(sl[c] - mx);
        float lse = mx + __logf(s);
        for (int c = 0; c < N_CLASSES; ++c)
            out[(size_t)gi * N_CLASSES + c] = sl[c] - lse;
    }
}

// --------------------------------------------------------------------- launch
extern "C" void kernel_launch(void* const* d_in, const int* in_sizes, int n_in,
                              void* d_out, int out_size, void* d_ws, size_t ws_size,
                              hipStream_t stream) {
    const float* x_in  = (const float*)d_in[0];
    // d_in[1] = edge_attr (ignored by SAGEConv)
    const int*   ei    = (const int*)d_in[2];
    const int*   batch = (const int*)d_in[3];
    const float* Wl    = (const float*)d_in[4];
    const float* bl    = (const float*)d_in[5];
    const float* Wr    = (const float*)d_in[6];
    const float* W1    = (const float*)d_in[7];
    const float* b1    = (const float*)d_in[8];
    const float* W2    = (const float*)d_in[9];
    const float* b2    = (const float*)d_in[10];
    float* out = (float*)d_out;

    // workspace carve-up (~52 MB)
    char* ws = (char*)d_ws;
    float* xw  = (float*)ws; ws += (size_t)N_NODES * HID * sizeof(float);
    float* msg = (float*)ws; ws += (size_t)N_NODES * HID * sizeof(float);
    float* cnt = (float*)ws; ws += (size_t)N_NODES * sizeof(float);   // counts -> inv counts
    float* g   = (float*)ws;

    const int NX = N_NODES * HID;                 // 6,400,000

    // working copy of node features (inputs must not be mutated)
    k_copy<<<(NX + 255) / 256, 256, 0, stream>>>(x_in, xw, NX);

    // degree counts (once), then invert in place
    k_zero<<<(N_NODES + 255) / 256, 256, 0, stream>>>(cnt, N_NODES);
    k_count<<<(N_EDGES + 255) / 256, 256, 0, stream>>>(ei, cnt);
    k_inv<<<(N_NODES + 255) / 256, 256, 0, stream>>>(cnt);

    for (int l = 0; l < N_LAYERS; ++l) {
        k_zero<<<(NX + 255) / 256, 256, 0, stream>>>(msg, NX);
        k_scatter<<<(N_EDGES * 32) / 256, 256, 0, stream>>>(xw, ei, msg);
        k_sage<<<N_NODES / 16, 256, 0, stream>>>(xw, msg, cnt,
                                                 Wl + (size_t)l * HID * HID,
                                                 bl + (size_t)l * HID,
                                                 Wr + (size_t)l * HID * HID);
    }

    k_zero<<<(N_GRAPHS * HID + 255) / 256, 256, 0, stream>>>(g, N_GRAPHS * HID);
    k_pool<<<NX / 256, 256, 0, stream>>>(xw, batch, g);
    k_head<<<N_GRAPHS, 128, 0, stream>>>(g, W1, b1, W2, b2, out);
}